// NeuralODE_53919019434088
// MI455X (gfx1250) — compile-verified
//
#include <hip/hip_runtime.h>

typedef __attribute__((ext_vector_type(2))) float v2f;
typedef __attribute__((ext_vector_type(8))) float v8f;

namespace {
constexpr int kBATCH = 2048, kD = 64, kW = 128, kT = 50;
constexpr int kWaves = 2;                       // waves (16-row tiles) per block
constexpr int kThreads = kWaves * 32;
constexpr int kBlocks = kBATCH / (kWaves * 16); // 64

// ---- dynamic-LDS layout (float element offsets) ----
constexpr int oW0K = 0;                 // [128][64]  w0[:,1:]
constexpr int oW1  = oW0K + kW * kD;    // [128][128] w1
constexpr int oW2  = oW1 + kW * kW;     // [64][128]  w2
constexpr int oW0T = oW2 + kD * kW;     // [128] w0[:,0]
constexpr int oB0  = oW0T + kW;
constexpr int oB1  = oB0 + kW;
constexpr int oB2  = oB1 + kW;
constexpr int oWS  = oB2 + kD;          // per-wave scratch base
constexpr int kWaveFloats = 12288;      // Y,YT,ACC,K1..K5 (8x1024) + H0,H1 (2x2048)
constexpr int oY = 0, oYT = 1024, oACC = 2048, oK1 = 3072, oK2 = 4096, oK3 = 5120,
              oK4 = 6144, oK5 = 7168, oH0 = 8192, oH1 = 10240;
constexpr int kLdsFloats = oWS + kWaves * kWaveFloats; // 57792
constexpr size_t kLdsBytes = kLdsFloats * sizeof(float); // 231168 B (< 320KB WGP LDS)

// Tsit5 tableau
constexpr float C2 = 0.161f, C3 = 0.327f, C4 = 0.9f, C5 = 0.9800255409045097f;
constexpr float A21 = 0.161f;
constexpr float A31 = -0.008480655492356989f, A32 = 0.335480655492357f;
constexpr float A41 = 2.8971530571054935f, A42 = -6.359448489975075f, A43 = 4.3622954328695815f;
constexpr float A51 = 5.325864828439257f, A52 = -11.748883564062828f,
                A53 = 7.4955393428898365f, A54 = -0.09249506636175525f;
constexpr float A61 = 5.86145544294642f, A62 = -12.92096931784711f, A63 = 8.159367898576159f,
                A64 = -0.071584973281401f, A65 = -0.028269050394068383f;
constexpr float B1 = 0.09646076681806523f, B2 = 0.01f, B3 = 0.4798896504144996f,
                B4 = 1.379008574103742f, B5 = -3.290069515436081f, B6 = 2.324710524099774f;
} // namespace

// fp32 tensor op: D(16x16,f32) = A(16x4,f32) x B(4x16,f32) + C
__device__ __forceinline__ v8f wmma4(v2f a, v2f b, v8f c) {
  return __builtin_amdgcn_wmma_f32_16x16x4_f32(false, a, false, b, (short)0, c, false, false);
}

// wave-local LDS ordering: drain own DS ops so cross-lane LDS reads see prior stores
__device__ __forceinline__ void wave_wait_lds() {
  asm volatile("s_wait_dscnt 0x0" ::: "memory");
}

// One dense layer on a 16-row tile: out[16,NT*16] = act(in[16,KD] @ w[N][K]^T + bias).
// All operands are addressed as integer offsets into the dynamic LDS block, which is
// declared *inside this function* with 16B alignment — address-space inference is
// purely local (everything lowers to ds_* ops) and the even-index v2f accesses are
// provably 8B-aligned so they lower to ds_load_b64 / ds_load_2addr_b64.
//   HAS_TW : fold t*tw[col] into the bias (layer 0 time column)
//   HAS_ACC: out = acc + accScale*result (fuses k6 into y_new)
//   RELU   : apply relu before store
template <int KD, int NT, bool HAS_TW, bool HAS_ACC, bool RELU>
__device__ __attribute__((noinline)) void mlp_layer(
    int inOff, int wOff, int biasOff, int twOff, float t,
    int outOff, int outStride, int accOff, float accScale, int lrow, int lhi) {
  extern __shared__ __align__(16) float smem[];
  v8f c[NT];
#pragma unroll
  for (int n = 0; n < NT; ++n) {
    const int col = n * 16 + lrow;
    float bv = smem[biasOff + col];
    if constexpr (HAS_TW) bv += t * smem[twOff + col];
    v8f cv = {bv, bv, bv, bv, bv, bv, bv, bv};
    c[n] = cv;
  }
#pragma unroll
  for (int kk = 0; kk < KD / 4; ++kk) {
    const int k0 = kk * 4 + 2 * lhi; // lanes 0-15 hold K={k0,k0+1}; lanes 16-31 K={k0+2,k0+3}
    const v2f a = *(const v2f*)&smem[inOff + lrow * KD + k0];
#pragma unroll
    for (int n = 0; n < NT; ++n) {
      const v2f b = *(const v2f*)&smem[wOff + (n * 16 + lrow) * KD + k0];
      c[n] = wmma4(a, b, c[n]);
    }
  }
#pragma unroll
  for (int n = 0; n < NT; ++n) {
    const int col = n * 16 + lrow;
#pragma unroll
    for (int v = 0; v < 8; ++v) {
      const int m = v + 8 * lhi; // C/D layout: vgpr v -> row v (lanes 0-15) / 8+v (lanes 16-31)
      float val = c[n][v];
      if constexpr (RELU) val = fmaxf(val, 0.0f);
      if constexpr (HAS_ACC) val = smem[accOff + m * outStride + col] + accScale * val;
      smem[outOff + m * outStride + col] = val;
    }
  }
}

__global__ __launch_bounds__(kThreads) void node_tsit5_kernel(
    const float* __restrict__ y0, const float* __restrict__ ts,
    const float* __restrict__ w0, const float* __restrict__ b0,
    const float* __restrict__ w1, const float* __restrict__ b1,
    const float* __restrict__ w2, const float* __restrict__ b2,
    float* __restrict__ out) {
  extern __shared__ __align__(16) float smem[];

  const int tid = threadIdx.x;
  // cooperative one-time weight load; weights stay LDS-resident for all 49 steps
  for (int i = tid; i < kW * kD; i += kThreads)
    smem[oW0K + i] = w0[(i >> 6) * (kD + 1) + 1 + (i & 63)];
  for (int i = tid; i < kW; i += kThreads) {
    smem[oW0T + i] = w0[i * (kD + 1)];
    smem[oB0 + i] = b0[i];
    smem[oB1 + i] = b1[i];
  }
  for (int i = tid; i < kW * kW; i += kThreads) smem[oW1 + i] = w1[i];
  for (int i = tid; i < kD * kW; i += kThreads) smem[oW2 + i] = w2[i];
  for (int i = tid; i < kD; i += kThreads) smem[oB2 + i] = b2[i];
  __syncthreads(); // the only workgroup barrier; waves are independent afterwards

  const int wave = tid >> 5;
  const int lid = tid & 31;
  const int lrow = lid & 15;
  const int lhi = lid >> 4;
  const int r0 = (blockIdx.x * kWaves + wave) * 16; // first batch row of this wave's tile

  const int WS = oWS + wave * kWaveFloats;
  const int Y = WS + oY, YT = WS + oYT, ACC = WS + oACC;
  const int K1 = WS + oK1, K2 = WS + oK2, K3 = WS + oK3, K4 = WS + oK4, K5 = WS + oK5;
  const int H0 = WS + oH0, H1 = WS + oH1;

  for (int i = lid; i < 16 * kD; i += 32) smem[Y + i] = y0[r0 * kD + i];
  const float dt = ts[1] - ts[0];
  wave_wait_lds();

  // f(t, y): 3-layer MLP entirely on the f32 matrix pipe (stages 1..5 path)
  auto dyn = [&](int yinOff, int koutOff, float teval) {
    mlp_layer<kD, 8, true, false, true>(yinOff, oW0K, oB0, oW0T, teval, H0, kW,
                                        oB0, 0.f, lrow, lhi);
    wave_wait_lds();
    mlp_layer<kW, 8, false, false, true>(H0, oW1, oB1, oB1, 0.f, H1, kW,
                                         oB1, 0.f, lrow, lhi);
    wave_wait_lds();
    mlp_layer<kW, 4, false, false, false>(H1, oW2, oB2, oB2, 0.f, koutOff, kD,
                                          oB2, 0.f, lrow, lhi);
    wave_wait_lds();
  };

#pragma unroll 1
  for (int step = 0; step < kT - 1; ++step) {
    const float t = ts[step];

    dyn(Y, K1, t);
    for (int i = lid; i < 1024; i += 32)
      smem[YT + i] = smem[Y + i] + dt * (A21 * smem[K1 + i]);
    wave_wait_lds();

    dyn(YT, K2, t + C2 * dt);
    for (int i = lid; i < 1024; i += 32)
      smem[YT + i] = smem[Y + i] + dt * (A31 * smem[K1 + i] + A32 * smem[K2 + i]);
    wave_wait_lds();

    dyn(YT, K3, t + C3 * dt);
    for (int i = lid; i < 1024; i += 32)
      smem[YT + i] = smem[Y + i] +
                     dt * (A41 * smem[K1 + i] + A42 * smem[K2 + i] + A43 * smem[K3 + i]);
    wave_wait_lds();

    dyn(YT, K4, t + C4 * dt);
    for (int i = lid; i < 1024; i += 32)
      smem[YT + i] = smem[Y + i] + dt * (A51 * smem[K1 + i] + A52 * smem[K2 + i] +
                                         A53 * smem[K3 + i] + A54 * smem[K4 + i]);
    wave_wait_lds();

    dyn(YT, K5, t + C5 * dt);
    for (int i = lid; i < 1024; i += 32) {
      const float k1 = smem[K1 + i], k2 = smem[K2 + i], k3 = smem[K3 + i],
                  k4 = smem[K4 + i], k5 = smem[K5 + i], y = smem[Y + i];
      smem[YT + i]  = y + dt * (A61 * k1 + A62 * k2 + A63 * k3 + A64 * k4 + A65 * k5);
      smem[ACC + i] = y + dt * (B1 * k1 + B2 * k2 + B3 * k3 + B4 * k4 + B5 * k5);
    }
    wave_wait_lds();

    // stage 6: same layer-0/1 instantiations, final layer fuses Y = ACC + dt*B6*k6
    mlp_layer<kD, 8, true, false, true>(YT, oW0K, oB0, oW0T, t + dt, H0, kW,
                                        oB0, 0.f, lrow, lhi);
    wave_wait_lds();
    mlp_layer<kW, 8, false, false, true>(H0, oW1, oB1, oB1, 0.f, H1, kW,
                                         oB1, 0.f, lrow, lhi);
    wave_wait_lds();
    mlp_layer<kW, 4, false, true, false>(H1, oW2, oB2, oB2, 0.f, Y, kD,
                                         ACC, dt * B6, lrow, lhi);
    wave_wait_lds();
  }

  for (int i = lid; i < 16 * kD; i += 32) out[r0 * kD + i] = smem[Y + i];
}

extern "C" void kernel_launch(void* const* d_in, const int* in_sizes, int n_in,
                              void* d_out, int out_size, void* d_ws, size_t ws_size,
                              hipStream_t stream) {
  const float* y0 = (const float*)d_in[0];
  const float* ts = (const float*)d_in[1];
  const float* w0 = (const float*)d_in[2];
  const float* b0 = (const float*)d_in[3];
  const float* w1 = (const float*)d_in[4];
  const float* b1 = (const float*)d_in[5];
  const float* w2 = (const float*)d_in[6];
  const float* b2 = (const float*)d_in[7];
  float* out = (float*)d_out;
  hipLaunchKernelGGL(node_tsit5_kernel, dim3(kBlocks), dim3(kThreads), kLdsBytes, stream,
                     y0, ts, w0, b0, w1, b1, w2, b2, out);
}